// DKVMN_53480932769794
// MI455X (gfx1250) — compile-verified
//
#include <hip/hip_runtime.h>
#include <hip/hip_bf16.h>

typedef __attribute__((ext_vector_type(2))) float v2f;
typedef __attribute__((ext_vector_type(8))) float v8f;

#define VOCAB    100
#define T_STEPS  128
#define BATCH    1024
#define MSLOT    32
#define DQ       64
#define DV       64
#define A_STRIDE 34   // padded LDS row stride (floats): 8B aligned, bank-conflict free

// preds[t,b] = c_t + sum_{s<t} (att_t . att_s) * r_s      (per batch element b)
//   A = softmax(Eq[q] @ Wa + ba)   (T x 32)
//   r_s = Ev[(2q+a)%100] . Wf_r ,  c_t = Eq[q_t] . Wf_q + bf
__global__ __launch_bounds__(128) void dkvmn_wmma_kernel(
    const int* __restrict__ questions, const int* __restrict__ answers,
    const float* __restrict__ Eq, const float* __restrict__ Ev,
    const float* __restrict__ Wa, const float* __restrict__ ba,
    const float* __restrict__ Wf, const float* __restrict__ bf,
    float* __restrict__ out)
{
    __shared__ float A_lds[T_STEPS * A_STRIDE]; // softmax attention rows
    __shared__ float r_lds[T_STEPS];            // qa_s . Wf_r
    __shared__ float pred_lds[T_STEPS];         // accumulator, init c_t

    const int b   = blockIdx.x;
    const int tid = threadIdx.x;

    // ---------------- Phase 1: per-timestep scalars + attention rows -------
    {
        const int t  = tid;                       // blockDim == T_STEPS == 128
        const int q  = questions[t * BATCH + b];
        const int an = answers[t * BATCH + b];
        const int wi = (q * 2 + an) % VOCAB;

        float logit[MSLOT];
        #pragma unroll
        for (int m = 0; m < MSLOT; ++m) logit[m] = ba[m];

        float c = bf[0];
        float r = 0.0f;
        for (int d = 0; d < DQ; ++d) {
            const float qe = Eq[q * DQ + d];
            c += qe * Wf[d];
            r += Ev[wi * DV + d] * Wf[DQ + d];
            #pragma unroll
            for (int m = 0; m < MSLOT; ++m)
                logit[m] = fmaf(qe, Wa[d * MSLOT + m], logit[m]);
        }

        // row softmax over M=32
        float mx = logit[0];
        #pragma unroll
        for (int m = 1; m < MSLOT; ++m) mx = fmaxf(mx, logit[m]);
        float se = 0.0f;
        #pragma unroll
        for (int m = 0; m < MSLOT; ++m) { logit[m] = __expf(logit[m] - mx); se += logit[m]; }
        const float inv = 1.0f / se;
        #pragma unroll
        for (int m = 0; m < MSLOT; ++m)
            A_lds[t * A_STRIDE + m] = logit[m] * inv;

        r_lds[t]    = r;
        pred_lds[t] = c;
    }
    __syncthreads();

    // ---------------- Phase 2: tril(A A^T) r via V_WMMA_F32_16X16X4_F32 ----
    // 8x8 grid of 16x16 S-tiles; 36 lower-triangular tiles, 9 per wave so
    // every wave runs WMMA unconditionally with EXEC all-ones.
    const int wave = tid >> 5;          // 0..3
    const int lane = tid & 31;
    const int mn   = lane & 15;         // A-row (M) / B-col (N) within tile
    const int hi   = lane >> 4;         // 0: lanes 0-15, 1: lanes 16-31
    const int koff = hi * 2;            // f32 A/B fragment K offset per ISA layout

    static const unsigned char TI[36] = {0,1,1,2,2,2,3,3,3,3,4,4,4,4,4,5,5,5,5,5,5,
                                         6,6,6,6,6,6,6,7,7,7,7,7,7,7,7};
    static const unsigned char TJ[36] = {0,0,1,0,1,2,0,1,2,3,0,1,2,3,4,0,1,2,3,4,5,
                                         0,1,2,3,4,5,6,0,1,2,3,4,5,6,7};

    for (int i = 0; i < 9; ++i) {
        const int idx = wave * 9 + i;   // uniform within a wave at runtime
        const int ti  = TI[idx];
        const int tj  = TJ[idx];

        const float* arow = &A_lds[(ti * 16 + mn) * A_STRIDE + koff];
        const float* brow = &A_lds[(tj * 16 + mn) * A_STRIDE + koff];

        v8f acc = {0.f, 0.f, 0.f, 0.f, 0.f, 0.f, 0.f, 0.f};
        #pragma unroll
        for (int k = 0; k < MSLOT; k += 4) {
            v2f a;  a.x  = arow[k]; a.y  = arow[k + 1];   // A: 16x4 f32 fragment
            v2f bb; bb.x = brow[k]; bb.y = brow[k + 1];   // B: 4x16 f32 fragment (A_tj^T)
            acc = __builtin_amdgcn_wmma_f32_16x16x4_f32(
                      false, a, false, bb, (short)0, acc, false, false);
        }

        // lane holds S[t, s] for s = tj*16 + mn, t = ti*16 + v + 8*hi.
        // Branchless causal mask ...
        const int   s  = tj * 16 + mn;
        const float rs = r_lds[s];

        float val[8];
        #pragma unroll
        for (int v = 0; v < 8; ++v) {
            const int t = ti * 16 + v + 8 * hi;
            val[v] = (s < t) ? acc[v] * rs : 0.0f;      // v_cndmask, no branch
        }

        // ... then a register-folding butterfly: sum 8 values across the 16
        // lanes of each half in 8 shuffles total (4+2+1+1), landing the sum
        // for accumulator v in lane (v) of each half — no final select tree.
        // step 1: xor 1, fold register pairs
        const float w0 = ((mn & 1) ? val[1] : val[0]) + __shfl_xor(((mn & 1) ? val[0] : val[1]), 1, 16);
        const float w1 = ((mn & 1) ? val[3] : val[2]) + __shfl_xor(((mn & 1) ? val[2] : val[3]), 1, 16);
        const float w2 = ((mn & 1) ? val[5] : val[4]) + __shfl_xor(((mn & 1) ? val[4] : val[5]), 1, 16);
        const float w3 = ((mn & 1) ? val[7] : val[6]) + __shfl_xor(((mn & 1) ? val[6] : val[7]), 1, 16);
        // step 2: xor 2
        const float x0 = ((mn & 2) ? w1 : w0) + __shfl_xor(((mn & 2) ? w0 : w1), 2, 16);
        const float x1 = ((mn & 2) ? w3 : w2) + __shfl_xor(((mn & 2) ? w2 : w3), 2, 16);
        // step 3: xor 4
        const float y  = ((mn & 4) ? x1 : x0) + __shfl_xor(((mn & 4) ? x0 : x1), 4, 16);
        // step 4: xor 8  ->  lane p of each half holds full sum of val[p & 7]
        const float z  = y + __shfl_xor(y, 8, 16);

        // lane mn (< 8) of each half commits row t = ti*16 + mn + 8*hi
        if (mn < 8)
            atomicAdd(&pred_lds[ti * 16 + mn + 8 * hi], z);
    }
    __syncthreads();

    // ---------------- Phase 3: write preds (T, B) --------------------------
    out[tid * BATCH + b] = pred_lds[tid];
}

extern "C" void kernel_launch(void* const* d_in, const int* in_sizes, int n_in,
                              void* d_out, int out_size, void* d_ws, size_t ws_size,
                              hipStream_t stream) {
    const int*   questions = (const int*)  d_in[0];
    const int*   answers   = (const int*)  d_in[1];
    const float* Eq        = (const float*)d_in[2];
    const float* Ev        = (const float*)d_in[3];
    const float* Wa        = (const float*)d_in[4];
    const float* ba        = (const float*)d_in[5];
    const float* Wf        = (const float*)d_in[6];
    const float* bf        = (const float*)d_in[7];
    float*       out       = (float*)d_out;

    dkvmn_wmma_kernel<<<dim3(BATCH), dim3(T_STEPS), 0, stream>>>(
        questions, answers, Eq, Ev, Wa, ba, Wf, bf, out);
}